// DFAGraphNet_7876970020891
// MI455X (gfx1250) — compile-verified
//
#include <hip/hip_runtime.h>
#include <hip/hip_bf16.h>
#include <math.h>

typedef float v2f __attribute__((ext_vector_type(2)));
typedef float v8f __attribute__((ext_vector_type(8)));

#define F_IN 512
#define HID  16
#define CLS  40
#define KC   128   // K-chunk staged in LDS per wave (16 x 128 floats = 8 KB)

// ---------------------------------------------------------------- init
__global__ void init_bufs(float* __restrict__ deg, float* __restrict__ agg1,
                          float* __restrict__ agg2, int N) {
    int i = blockIdx.x * blockDim.x + threadIdx.x;
    if (i < N)        deg[i]  = 1.0f;   // self-loop weight
    if (i < N * HID)  agg1[i] = 0.0f;
    if (i < N * CLS)  agg2[i] = 0.0f;
}

// ---------------------------------------------------------------- degree
__global__ void deg_edges(const long long* __restrict__ col,
                          const float* __restrict__ ew,
                          float* __restrict__ deg, int E) {
    int e = blockIdx.x * blockDim.x + threadIdx.x;
    if (e < E) atomicAdd(&deg[(int)col[e]], ew[e]);
}

__global__ void dinv_k(float* __restrict__ deg, int N) {
    int i = blockIdx.x * blockDim.x + threadIdx.x;
    if (i < N) {
        float d = deg[i];
        deg[i] = (d > 0.0f) ? rsqrtf(d) : 0.0f;   // deg becomes dinv in place
    }
}

__global__ void norm_k(const long long* __restrict__ row,
                       const long long* __restrict__ col,
                       const float* __restrict__ ew,
                       const float* __restrict__ dinv,
                       float* __restrict__ norm, int E) {
    int e = blockIdx.x * blockDim.x + threadIdx.x;
    if (e < E) norm[e] = dinv[(int)row[e]] * ew[e] * dinv[(int)col[e]];
}

// ---------------------------------------------------------------- GEMM1: h1 = x @ W1  (N x 512 @ 512 x 16)
// One wave per 16-row tile. K staged through LDS in 16x128 chunks (8 KB/wave
// -> ~40 waves/WGP LDS-limited) with coalesced b128 global loads; A fragments
// consumed via ds_load in WMMA f32 16x16x4 layout. Two accumulators break the
// WMMA RAW chain.
__global__ void __launch_bounds__(64) gemm1_wmma(const float* __restrict__ x,
                                                 const float* __restrict__ W1,
                                                 float* __restrict__ h1,
                                                 int ntiles) {
    __shared__ float lds[2][16 * KC];   // 16 KB per 2-wave block
    const int wave = threadIdx.x >> 5;
    const int lane = threadIdx.x & 31;
    const int tile = blockIdx.x * 2 + wave;
    if (tile >= ntiles) return;

    float* tl = lds[wave];
    const float* xt = x + (size_t)tile * 16 * F_IN;

    const int m     = lane & 15;       // row within tile / column of B
    const int khalf = lane >> 4;       // selects K pair per A/B fragment layout

    v8f acc0 = {}, acc1 = {};

    for (int kc = 0; kc < F_IN; kc += KC) {
        // cooperative coalesced stage of the 16 x KC chunk:
        // 16*KC/4 = 512 float4, 16 per lane
        #pragma unroll 4
        for (int i = 0; i < (16 * KC) / (4 * 32); ++i) {
            int idx4 = lane + 32 * i;          // 0..511
            int r    = idx4 >> 5;              // KC/4 = 32 float4 per row
            int c4   = idx4 & 31;
            ((float4*)tl)[idx4] =
                *(const float4*)(xt + (size_t)r * F_IN + kc + 4 * c4);
        }
        asm volatile("s_wait_dscnt 0" ::: "memory");   // cross-lane LDS visibility

        #pragma unroll 4
        for (int k = 0; k < KC; k += 8) {
            const int kk = k + 2 * khalf;
            v2f a0 = *(const v2f*)&tl[m * KC + kk];        // A frags (16x4 f32)
            v2f a1 = *(const v2f*)&tl[m * KC + kk + 4];
            v2f b0, b1;                                    // B frags (4x16 f32)
            b0.x = W1[(kc + kk) * HID + m];
            b0.y = W1[(kc + kk + 1) * HID + m];
            b1.x = W1[(kc + kk + 4) * HID + m];
            b1.y = W1[(kc + kk + 5) * HID + m];
            acc0 = __builtin_amdgcn_wmma_f32_16x16x4_f32(
                false, a0, false, b0, (short)0, acc0, false, false);
            acc1 = __builtin_amdgcn_wmma_f32_16x16x4_f32(
                false, a1, false, b1, (short)0, acc1, false, false);
        }
    }

    v8f acc = acc0 + acc1;
    // C/D layout: VGPR v holds row m0 + v + 8*khalf, column m
    float* out = h1 + ((size_t)tile * 16 + 8 * khalf) * HID + m;
    #pragma unroll
    for (int v = 0; v < 8; ++v) out[(size_t)v * HID] = acc[v];
}

// ---------------------------------------------------------------- scatter layer 1
__global__ void scatter1(const long long* __restrict__ row,
                         const long long* __restrict__ col,
                         const float* __restrict__ norm,
                         const float* __restrict__ h1,
                         float* __restrict__ agg1, int E) {
    int idx = blockIdx.x * blockDim.x + threadIdx.x;   // E*16 < 2^31
    if (idx >= E * HID) return;
    int e = idx >> 4, f = idx & 15;
    int r = (int)row[e], c = (int)col[e];
    float v = norm[e] * h1[(size_t)r * HID + f];
    atomicAdd(&agg1[(size_t)c * HID + f], v);
}

// ---------------------------------------------------------------- self-loop + bias + ReLU (z1 in place of agg1)
__global__ void relu_bias(float* __restrict__ agg1,
                          const float* __restrict__ h1,
                          const float* __restrict__ dinv,
                          const float* __restrict__ b1, int N) {
    int idx = blockIdx.x * blockDim.x + threadIdx.x;
    if (idx >= N * HID) return;
    int i = idx >> 4, f = idx & 15;
    float di = dinv[i];
    float v = agg1[idx] + di * di * h1[idx] + b1[f];
    agg1[idx] = fmaxf(v, 0.0f);
}

// ---------------------------------------------------------------- GEMM2: h2 = z1 @ W2  (N x 16 @ 16 x 40)
__global__ void __launch_bounds__(64) gemm2_wmma(const float* __restrict__ z1,
                                                 const float* __restrict__ W2,
                                                 float* __restrict__ h2,
                                                 int ntiles) {
    const int wave = threadIdx.x >> 5;
    const int lane = threadIdx.x & 31;
    const int tile = blockIdx.x * 2 + wave;
    if (tile >= ntiles) return;

    const int m     = lane & 15;
    const int khalf = lane >> 4;
    const float* zrow = z1 + ((size_t)tile * 16 + m) * HID;

    v8f acc0 = {}, acc1 = {}, acc2 = {};
    #pragma unroll
    for (int k = 0; k < HID; k += 4) {
        const int kk = k + 2 * khalf;
        v2f a = *(const v2f*)(zrow + kk);
        v2f b0, b1v, b2v;
        b0.x  = W2[kk * CLS + m];            b0.y  = W2[(kk + 1) * CLS + m];
        b1v.x = W2[kk * CLS + 16 + m];       b1v.y = W2[(kk + 1) * CLS + 16 + m];
        b2v.x = (32 + m < CLS) ? W2[kk * CLS + 32 + m] : 0.0f;
        b2v.y = (32 + m < CLS) ? W2[(kk + 1) * CLS + 32 + m] : 0.0f;
        acc0 = __builtin_amdgcn_wmma_f32_16x16x4_f32(false, a, false, b0,  (short)0, acc0, false, false);
        acc1 = __builtin_amdgcn_wmma_f32_16x16x4_f32(false, a, false, b1v, (short)0, acc1, false, false);
        acc2 = __builtin_amdgcn_wmma_f32_16x16x4_f32(false, a, false, b2v, (short)0, acc2, false, false);
    }

    #pragma unroll
    for (int v = 0; v < 8; ++v) {
        size_t r = (size_t)tile * 16 + v + 8 * khalf;
        h2[r * CLS + m]      = acc0[v];
        h2[r * CLS + 16 + m] = acc1[v];
        if (32 + m < CLS) h2[r * CLS + 32 + m] = acc2[v];
    }
}

// ---------------------------------------------------------------- scatter layer 2
__global__ void scatter2(const long long* __restrict__ row,
                         const long long* __restrict__ col,
                         const float* __restrict__ norm,
                         const float* __restrict__ h2,
                         float* __restrict__ agg2, int E) {
    int idx = blockIdx.x * blockDim.x + threadIdx.x;   // E*40 < 2^31
    if (idx >= E * CLS) return;
    int e = idx / CLS, f = idx % CLS;
    int r = (int)row[e], c = (int)col[e];
    float v = norm[e] * h2[(size_t)r * CLS + f];
    atomicAdd(&agg2[(size_t)c * CLS + f], v);
}

// ---------------------------------------------------------------- self-loop + bias + log_softmax
__global__ void finalize(const float* __restrict__ agg2,
                         const float* __restrict__ h2,
                         const float* __restrict__ dinv,
                         const float* __restrict__ b2,
                         float* __restrict__ out, int N) {
    int i = blockIdx.x * blockDim.x + threadIdx.x;
    if (i >= N) return;
    float di = dinv[i];
    float d2 = di * di;
    float l[CLS];
    float mx = -3.402823466e38f;
    #pragma unroll
    for (int f = 0; f < CLS; ++f) {
        float v = agg2[(size_t)i * CLS + f] + d2 * h2[(size_t)i * CLS + f] + b2[f];
        l[f] = v;
        mx = fmaxf(mx, v);
    }
    float s = 0.0f;
    #pragma unroll
    for (int f = 0; f < CLS; ++f) s += expf(l[f] - mx);
    float lse = mx + logf(s);
    #pragma unroll
    for (int f = 0; f < CLS; ++f) out[(size_t)i * CLS + f] = l[f] - lse;
}

// ---------------------------------------------------------------- launch
extern "C" void kernel_launch(void* const* d_in, const int* in_sizes, int n_in,
                              void* d_out, int out_size, void* d_ws, size_t ws_size,
                              hipStream_t stream) {
    const float*     x  = (const float*)d_in[0];
    const long long* ei = (const long long*)d_in[1];   // int64 [2, E]
    const float*     ew = (const float*)d_in[2];
    const float*     W1 = (const float*)d_in[3];
    const float*     b1 = (const float*)d_in[4];
    const float*     W2 = (const float*)d_in[5];
    const float*     b2 = (const float*)d_in[6];

    const int N = in_sizes[0] / F_IN;   // 100000 (divisible by 16)
    const int E = in_sizes[2];          // 3200000
    const long long* row = ei;
    const long long* col = ei + E;

    // workspace layout (floats)
    float* ws   = (float*)d_ws;
    float* deg  = ws;                        // N   (becomes dinv in place)
    float* h1   = deg  + N;                  // N*16
    float* agg1 = h1   + (size_t)N * HID;    // N*16 (becomes z1 in place)
    float* h2   = agg1 + (size_t)N * HID;    // N*40
    float* agg2 = h2   + (size_t)N * CLS;    // N*40
    float* norm = agg2 + (size_t)N * CLS;    // E

    const int T = 256;
    const int ntiles = N / 16;

    init_bufs<<<((size_t)N * CLS + T - 1) / T, T, 0, stream>>>(deg, agg1, agg2, N);
    deg_edges<<<(E + T - 1) / T, T, 0, stream>>>(col, ew, deg, E);
    dinv_k<<<(N + T - 1) / T, T, 0, stream>>>(deg, N);
    norm_k<<<(E + T - 1) / T, T, 0, stream>>>(row, col, ew, deg, norm, E);

    gemm1_wmma<<<(ntiles + 1) / 2, 64, 0, stream>>>(x, W1, h1, ntiles);
    scatter1<<<((size_t)E * HID + T - 1) / T, T, 0, stream>>>(row, col, norm, h1, agg1, E);
    relu_bias<<<((size_t)N * HID + T - 1) / T, T, 0, stream>>>(agg1, h1, deg, b1, N);

    gemm2_wmma<<<(ntiles + 1) / 2, 64, 0, stream>>>(agg1, W2, h2, ntiles);
    scatter2<<<((size_t)E * CLS + T - 1) / T, T, 0, stream>>>(row, col, norm, h2, agg2, E);
    finalize<<<(N + T - 1) / T, T, 0, stream>>>(agg2, h2, deg, b2, (float*)d_out, N);
}